// LDet_28561532518420
// MI455X (gfx1250) — compile-verified
//
#include <hip/hip_runtime.h>
#include <hip/hip_bf16.h>
#include <float.h>
#include <math.h>

#define N_ANCH 21824
#define NTILES 1364      // N_ANCH / 16, exact
#define MGT 32
#define NCLS 20
#define NBINS 16
#define NB 62            // stage-1 blocks per image
#define TPB 22           // tiles per stage-1 block: 62*22 == 1364 exactly

typedef float v2f __attribute__((ext_vector_type(2)));
typedef float v8f __attribute__((ext_vector_type(8)));

// Sortable key: (sign-flipped float bits << 32) | anchor index.
// u64 ascending order == (dist ascending, then index ascending) — matches
// jax.lax.top_k tie-breaking (lowest index wins among equal distances).
__device__ __forceinline__ unsigned int fkey(float f) {
    unsigned int u = __float_as_uint(f);
    unsigned int m = (unsigned int)((int)u >> 31);
    return u ^ (m | 0x80000000u);
}
__device__ __forceinline__ unsigned long long mkkey(float d, int idx) {
    return ((unsigned long long)fkey(d) << 32) | (unsigned int)idx;
}

// Branch-free insert into a sorted-ascending 9-key list held in registers.
__device__ __forceinline__ void ins9k(unsigned long long kk, unsigned long long (&K)[9]) {
    const bool rep = kk < K[8];
    K[8] = rep ? kk : K[8];
    #pragma unroll
    for (int k = 8; k > 0; --k) {
        const bool sw = K[k] < K[k-1];
        const unsigned long long a = K[k-1], b = K[k];
        K[k-1] = sw ? b : a;
        K[k]   = sw ? a : b;
    }
}

// ---------------------------------------------------------------------------
// Stage 1: per (image, 22-tile chunk): WMMA pairwise dist^2, per-lane top-9,
// LDS merge -> sorted top-9 keys per gt written to workspace.
//   dist^2[n,m] = [acx, acy, acx^2+acy^2, 1] . [-2gcx, -2gcy, 1, gcx^2+gcy^2]
// A layout (32b 16x4): v0 = K0 (lanes 0-15) / K2 (lanes 16-31); v1 = K1 / K3.
// ---------------------------------------------------------------------------
__global__ __launch_bounds__(256) void atss_part1_kernel(
    const float* __restrict__ anchors,
    const float* __restrict__ gt_boxes,
    unsigned long long* __restrict__ cand_out)   // [B][NB][MGT][9]
{
    const int b    = blockIdx.y;
    const int blk  = blockIdx.x;
    const int tid  = threadIdx.x;
    const int wave = tid >> 5;
    const int lane = tid & 31;
    const int col  = lane & 15;
    const bool hi  = lane >= 16;
    const int t0   = blk * TPB;

    __shared__ float s_gt[MGT * 4];
    __shared__ unsigned long long s_cand[MGT * 144];   // 16 slots * 9 per gt

    for (int i = tid; i < MGT * 4; i += 256)
        s_gt[i] = gt_boxes[(size_t)b * MGT * 4 + i];
    __syncthreads();

    const int g0 = col, g1 = 16 + col;
    const float g0cx = 0.5f * (s_gt[g0*4+0] + s_gt[g0*4+2]);
    const float g0cy = 0.5f * (s_gt[g0*4+1] + s_gt[g0*4+3]);
    const float g1cx = 0.5f * (s_gt[g1*4+0] + s_gt[g1*4+2]);
    const float g1cy = 0.5f * (s_gt[g1*4+1] + s_gt[g1*4+3]);

    v2f B0, B1;
    if (!hi) {
        B0[0] = -2.0f * g0cx; B0[1] = -2.0f * g0cy;
        B1[0] = -2.0f * g1cx; B1[1] = -2.0f * g1cy;
    } else {
        B0[0] = 1.0f; B0[1] = g0cx*g0cx + g0cy*g0cy;
        B1[0] = 1.0f; B1[1] = g1cx*g1cx + g1cy*g1cy;
    }

    unsigned long long KA[9], KB[9];
    #pragma unroll
    for (int k = 0; k < 9; ++k) { KA[k] = ~0ull; KB[k] = ~0ull; }

    for (int t = t0 + wave; t < t0 + TPB; t += 8) {  // wave-uniform: EXEC==all-1 at WMMA
        const int row = t * 16 + col;
        const float a0 = anchors[row*4+0], a1 = anchors[row*4+1];
        const float a2 = anchors[row*4+2], a3 = anchors[row*4+3];
        const float acx = 0.5f * (a0 + a2), acy = 0.5f * (a1 + a3);
        v2f A;
        if (!hi) { A[0] = acx; A[1] = acy; }
        else     { A[0] = acx*acx + acy*acy; A[1] = 1.0f; }
        v8f c = {};
        v8f D0 = __builtin_amdgcn_wmma_f32_16x16x4_f32(false, A, false, B0, (short)0, c, false, false);
        v8f D1 = __builtin_amdgcn_wmma_f32_16x16x4_f32(false, A, false, B1, (short)0, c, false, false);
        const int rbase = t * 16 + (hi ? 8 : 0);
        #pragma unroll
        for (int v = 0; v < 8; ++v) {
            ins9k(mkkey(D0[v], rbase + v), KA);
            ins9k(mkkey(D1[v], rbase + v), KB);
        }
    }

    const int slot = wave * 2 + (hi ? 1 : 0);   // 0..15
    #pragma unroll
    for (int k = 0; k < 9; ++k) {
        s_cand[g0*144 + slot*9 + k] = KA[k];
        s_cand[g1*144 + slot*9 + k] = KB[k];
    }
    __syncthreads();

    if (tid < MGT) {
        const int g = tid;
        unsigned long long prev = 0ull;   // all real keys are > 0
        #pragma unroll
        for (int r = 0; r < 9; ++r) {
            unsigned long long best = ~0ull;
            for (int j = 0; j < 144; ++j) {
                const unsigned long long k = s_cand[g*144 + j];
                if (k > prev && k < best) best = k;
            }
            cand_out[(((size_t)b * NB + blk) * MGT + g) * 9 + r] = best;
            prev = best;
        }
    }
}

// ---------------------------------------------------------------------------
// Stage 2: merge NB sorted top-9 lists per (image, gt) -> global top-9 ->
// IoUs of the 9 closest anchors -> thr = mean + std(ddof=1).
// ---------------------------------------------------------------------------
__global__ __launch_bounds__(32) void atss_part2_kernel(
    const float* __restrict__ anchors,
    const float* __restrict__ gt_boxes,
    const unsigned long long* __restrict__ cand,   // [B][NB][MGT][9]
    float* __restrict__ thr_out)
{
    const int b = blockIdx.x;
    const int g = threadIdx.x;     // 32 threads: one gt each

    const float gx0 = gt_boxes[((size_t)b*MGT + g)*4 + 0];
    const float gy0 = gt_boxes[((size_t)b*MGT + g)*4 + 1];
    const float gx2 = gt_boxes[((size_t)b*MGT + g)*4 + 2];
    const float gy2 = gt_boxes[((size_t)b*MGT + g)*4 + 3];
    const float areag = (gx2 - gx0) * (gy2 - gy0);

    float vals[9]; float sum = 0.0f;
    unsigned long long prev = 0ull;
    #pragma unroll
    for (int r = 0; r < 9; ++r) {
        unsigned long long best = ~0ull;
        for (int q = 0; q < NB; ++q) {
            const size_t base = (((size_t)b * NB + q) * MGT + g) * 9;
            #pragma unroll
            for (int r2 = 0; r2 < 9; ++r2) {
                const unsigned long long k = cand[base + r2];
                if (k > prev && k < best) best = k;
            }
        }
        prev = best;
        const int ai = (int)(best & 0xFFFFFFFFull);
        const float a0 = anchors[ai*4+0], a1 = anchors[ai*4+1];
        const float a2 = anchors[ai*4+2], a3 = anchors[ai*4+3];
        const float areaa = (a2 - a0) * (a3 - a1);
        const float lx = fmaxf(a0, gx0), ly = fmaxf(a1, gy0);
        const float rx = fminf(a2, gx2), ry = fminf(a3, gy2);
        const float iw = fmaxf(rx - lx, 0.0f), ih = fmaxf(ry - ly, 0.0f);
        const float inter = iw * ih;
        const float iou = inter / fmaxf(areaa + areag - inter, 1e-9f);
        vals[r] = iou; sum += iou;
    }
    const float mean = sum * (1.0f / 9.0f);
    float var = 0.0f;
    #pragma unroll
    for (int r = 0; r < 9; ++r) { const float d = vals[r] - mean; var += d * d; }
    var *= (1.0f / 8.0f);                 // ddof=1
    thr_out[b * MGT + g] = mean + sqrtf(var);
}

// ---------------------------------------------------------------------------
// Kernel B: per-anchor matching + fused QFL/DFL/decode/GIoU, block partials.
// ---------------------------------------------------------------------------
__global__ __launch_bounds__(256) void loss_main_kernel(
    const float* __restrict__ cls_preds,
    const float* __restrict__ reg_preds,
    const float* __restrict__ anchors,
    const float* __restrict__ gt_boxes,
    const int*   __restrict__ gt_labels,
    const float* __restrict__ thr,
    float* __restrict__ partials)
{
    const int b   = blockIdx.y;
    const int tid = threadIdx.x;
    const int n   = blockIdx.x * 256 + tid;

    __shared__ float s_gt[MGT * 4];
    __shared__ int   s_lab[MGT];
    __shared__ float s_thr[MGT];
    if (tid < MGT * 4) s_gt[tid] = gt_boxes[(size_t)b * MGT * 4 + tid];
    if (tid >= 128 && tid < 128 + MGT) s_lab[tid - 128] = gt_labels[b * MGT + (tid - 128)];
    if (tid >= 160 && tid < 160 + MGT) s_thr[tid - 160] = thr[b * MGT + (tid - 160)];
    __syncthreads();

    float qfl = 0.0f, dflv = 0.0f, giouv = 0.0f, posf = 0.0f;

    if (n < N_ANCH) {
        const float a0 = anchors[n*4+0], a1 = anchors[n*4+1];
        const float a2 = anchors[n*4+2], a3 = anchors[n*4+3];
        const float acx = 0.5f * (a0 + a2), acy = 0.5f * (a1 + a3);
        const float areaa = (a2 - a0) * (a3 - a1);

        int matched = -1; float sel = 0.0f;
        #pragma unroll 4
        for (int m = 0; m < MGT; ++m) {   // sequential: last positive gt wins
            const float gx0 = s_gt[m*4+0], gy0 = s_gt[m*4+1];
            const float gx2 = s_gt[m*4+2], gy2 = s_gt[m*4+3];
            const float areag = (gx2 - gx0) * (gy2 - gy0);
            const float lx = fmaxf(a0, gx0), ly = fmaxf(a1, gy0);
            const float rx = fminf(a2, gx2), ry = fminf(a3, gy2);
            const float iw = fmaxf(rx - lx, 0.0f), ih = fmaxf(ry - ly, 0.0f);
            const float inter = iw * ih;
            const float iou = inter / fmaxf(areaa + areag - inter, 1e-9f);
            const bool inside = (acx >= gx0) && (acx <= gx2) && (acy >= gy0) && (acy <= gy2);
            const bool hit = inside && (iou >= s_thr[m]);
            matched = hit ? m : matched;
            sel     = hit ? iou : sel;
        }
        const bool  pos  = matched >= 0;
        const int   clst = pos ? s_lab[matched] : 0;
        const float iout = pos ? sel : 0.0f;
        posf = pos ? 1.0f : 0.0f;          // labels >= 1 so (cls_t>0) == pos

        // ---- Quality Focal Loss ----
        const float* cp = cls_preds + ((size_t)b * N_ANCH + n) * NCLS;
        #pragma unroll 4
        for (int c = 0; c < NCLS; ++c) {
            const float x  = cp[c];
            const float oh = (c == clst) ? 1.0f : 0.0f;
            const float ps = 1.0f / (1.0f + expf(-x));
            const float pt = oh * ps + (1.0f - oh) * (1.0f - ps);
            float w = iout * (1.0f - pt) + (1.0f - iout) * pt;
            w = w * w;
            const float bce = fmaxf(x, 0.0f) - x * oh + log1pf(expf(-fabsf(x)));
            qfl += w * bce;
        }

        // ---- DFL + softmax-expectation decode ----
        float rt[4];
        #pragma unroll
        for (int s = 0; s < 4; ++s) rt[s] = pos ? s_gt[matched*4+s] : 0.0f;

        const float* rp = reg_preds + ((size_t)b * N_ANCH + n) * (4 * NBINS);
        float deltas[4];
        #pragma unroll
        for (int s = 0; s < 4; ++s) {
            const float* ls = rp + s * NBINS;
            float mx = -FLT_MAX;
            #pragma unroll
            for (int k = 0; k < NBINS; ++k) mx = fmaxf(mx, ls[k]);
            const float ts = rt[s] * (float)(NBINS - 1);
            int li = (int)ts; li = li < 0 ? 0 : (li > NBINS - 2 ? NBINS - 2 : li);
            float se = 0.0f, num = 0.0f, lli = 0.0f, lri = 0.0f;
            #pragma unroll
            for (int k = 0; k < NBINS; ++k) {
                const float lv = ls[k];
                const float e = expf(lv - mx);
                se  += e;
                num += e * ((float)k * (1.0f / (float)(NBINS - 1)));
                if (k == li)     lli = lv;
                if (k == li + 1) lri = lv;
            }
            const float logZ = mx + logf(se);
            deltas[s] = num / se;
            if (pos) {
                const float wr = ts - (float)li;
                const float wl = 1.0f - wr;
                dflv += -(wl * (lli - logZ) + wr * (lri - logZ));
            }
        }

        // ---- delta2bbox + GIoU ----
        if (pos) {
            const float w  = a2 - a0, h = a3 - a1;
            const float pcx = deltas[0] * w + acx, pcy = deltas[1] * h + acy;
            const float pw  = expf(deltas[2]) * w, ph = expf(deltas[3]) * h;
            const float p0 = pcx - 0.5f * pw, p1 = pcy - 0.5f * ph;
            const float p2 = pcx + 0.5f * pw, p3 = pcy + 0.5f * ph;
            const float lx = fmaxf(p0, rt[0]), ly = fmaxf(p1, rt[1]);
            const float rx = fminf(p2, rt[2]), ry = fminf(p3, rt[3]);
            const float iw = fmaxf(rx - lx, 0.0f), ih = fmaxf(ry - ly, 0.0f);
            const float inter = iw * ih;
            const float aa = (p2 - p0) * (p3 - p1);
            const float ab = (rt[2] - rt[0]) * (rt[3] - rt[1]);
            const float uni = aa + ab - inter;
            const float iou = inter / fmaxf(uni, 1e-9f);
            const float ex0 = fminf(p0, rt[0]), ey0 = fminf(p1, rt[1]);
            const float ex1 = fmaxf(p2, rt[2]), ey1 = fmaxf(p3, rt[3]);
            const float ew = fmaxf(ex1 - ex0, 0.0f), eh = fmaxf(ey1 - ey0, 0.0f);
            const float enc = ew * eh;
            const float giou = iou - (enc - uni) / fmaxf(enc, 1e-9f);
            giouv = 1.0f - giou;
        }
    }

    // ---- deterministic block reduction ----
    __shared__ float r0[256], r1[256], r2[256], r3[256];
    r0[tid] = qfl; r1[tid] = posf; r2[tid] = dflv; r3[tid] = giouv;
    __syncthreads();
    for (int off = 128; off > 0; off >>= 1) {
        if (tid < off) {
            r0[tid] += r0[tid + off]; r1[tid] += r1[tid + off];
            r2[tid] += r2[tid + off]; r3[tid] += r3[tid + off];
        }
        __syncthreads();
    }
    if (tid == 0) {
        const int bId = blockIdx.y * gridDim.x + blockIdx.x;
        partials[bId*4+0] = r0[0]; partials[bId*4+1] = r1[0];
        partials[bId*4+2] = r2[0]; partials[bId*4+3] = r3[0];
    }
}

// ---------------------------------------------------------------------------
// Kernel C: deterministic final reduction + loss normalization.
// ---------------------------------------------------------------------------
__global__ __launch_bounds__(256) void finalize_kernel(
    const float* __restrict__ partials, int nblocks, float* __restrict__ out)
{
    __shared__ float s0[256], s1[256], s2[256], s3[256];
    const int t = threadIdx.x;
    float a0 = 0.0f, a1 = 0.0f, a2 = 0.0f, a3 = 0.0f;
    for (int i = t; i < nblocks; i += 256) {
        a0 += partials[i*4+0]; a1 += partials[i*4+1];
        a2 += partials[i*4+2]; a3 += partials[i*4+3];
    }
    s0[t] = a0; s1[t] = a1; s2[t] = a2; s3[t] = a3;
    __syncthreads();
    for (int off = 128; off > 0; off >>= 1) {
        if (t < off) {
            s0[t] += s0[t + off]; s1[t] += s1[t + off];
            s2[t] += s2[t + off]; s3[t] += s3[t + off];
        }
        __syncthreads();
    }
    if (t == 0) {
        const float qfl_sum  = s0[0];
        const float pos_sum  = s1[0];
        const float dfl_sum  = s2[0];
        const float giou_sum = s3[0];
        const float loss_qfl  = qfl_sum / fmaxf(pos_sum, 1.0f);
        float loss_dfl = dfl_sum / fmaxf(pos_sum * 4.0f, 1.0f);
        loss_dfl = fminf(fmaxf(loss_dfl, 0.0f), 1.0f);
        const float loss_giou = giou_sum / fmaxf(pos_sum, 1.0f);
        out[0] = loss_qfl + loss_dfl + loss_giou;
        out[1] = loss_qfl;
        out[2] = loss_dfl;
        out[3] = loss_giou;
    }
}

extern "C" void kernel_launch(void* const* d_in, const int* in_sizes, int n_in,
                              void* d_out, int out_size, void* d_ws, size_t ws_size,
                              hipStream_t stream) {
    (void)in_sizes; (void)n_in; (void)out_size; (void)ws_size;
    const float* cls = (const float*)d_in[0];
    const float* reg = (const float*)d_in[1];
    const float* anc = (const float*)d_in[2];
    const float* gtb = (const float*)d_in[3];
    const int*   gtl = (const int*)d_in[4];
    float* out = (float*)d_out;

    const int Bimg = 16;
    // workspace layout
    float* thr = (float*)d_ws;                                            // 512 floats
    unsigned long long* cand = (unsigned long long*)((char*)d_ws + 4096); // B*NB*MGT*9 u64 = ~2.29 MB
    const size_t cand_bytes = (size_t)Bimg * NB * MGT * 9 * sizeof(unsigned long long);
    float* partials = (float*)((char*)d_ws + 4096 + ((cand_bytes + 4095) & ~(size_t)4095));

    atss_part1_kernel<<<dim3(NB, Bimg), dim3(256), 0, stream>>>(anc, gtb, cand);
    atss_part2_kernel<<<dim3(Bimg), dim3(32), 0, stream>>>(anc, gtb, cand, thr);

    dim3 grid((N_ANCH + 255) / 256, Bimg);                                // 86 x 16
    loss_main_kernel<<<grid, dim3(256), 0, stream>>>(cls, reg, anc, gtb, gtl, thr, partials);

    finalize_kernel<<<dim3(1), dim3(256), 0, stream>>>(partials, (int)(grid.x * Bimg), out);
}